// RotatedIoULoss_3994319586159
// MI455X (gfx1250) — compile-verified
//
#include <hip/hip_runtime.h>
#include <math.h>

#define LOSS_EPS 1e-6f
#define DEN_EPS  1e-10f
#define IN_TOL   1e-5f

typedef float v2f __attribute__((ext_vector_type(2)));
typedef float v8f __attribute__((ext_vector_type(8)));

// ---------------------------------------------------------------------------
// Compile-time Batcher odd-even mergesort network for 24 elements, generated
// by building the n=32 network and pruning comparators whose high index >= 24
// (valid by the +inf-padding argument: such comparators are provable no-ops).
// All indices are constexpr so the sorted arrays stay in VGPRs.
// ---------------------------------------------------------------------------
struct SortNet {
  int lo[192];
  int hi[192];
  int cnt;
};

constexpr SortNet make_sort_net() {
  SortNet net{};
  net.cnt = 0;
  const int n = 32, lim = 24;
  for (int p = 1; p < n; p <<= 1)
    for (int k = p; k >= 1; k >>= 1)
      for (int j = k % p; j + k <= n - 1; j += 2 * k)
        for (int i = 0; i <= k - 1 && i + j + k <= n - 1; ++i)
          if ((i + j) / (2 * p) == (i + j + k) / (2 * p))
            if (i + j + k < lim) {
              net.lo[net.cnt] = i + j;
              net.hi[net.cnt] = i + j + k;
              ++net.cnt;
            }
  return net;
}

constexpr SortNet SORT_NET = make_sort_net();

// ---------------------------------------------------------------------------
// Kernel 1: per-pair rotated IoU loss + deterministic per-block partial sums.
// One thread = one (pred, target) box pair; everything register-resident.
// ---------------------------------------------------------------------------
__global__ __launch_bounds__(256) void rot_iou_loss_kernel(
    const float* __restrict__ pred, const float* __restrict__ target,
    float* __restrict__ partial, int n) {
  const int i = blockIdx.x * 256 + threadIdx.x;

  float loss = 0.0f;
  if (i < n) {
    const float px = pred[i * 5 + 0], py = pred[i * 5 + 1];
    const float pw = pred[i * 5 + 2], ph = pred[i * 5 + 3];
    const float pa = pred[i * 5 + 4];
    const float qx = target[i * 5 + 0], qy = target[i * 5 + 1];
    const float qw = target[i * 5 + 2], qh = target[i * 5 + 3];
    const float qa = target[i * 5 + 4];

    float s1, c1a, s2, c2a;
    sincosf(pa, &s1, &c1a);
    sincosf(qa, &s2, &c2a);

    const float dxs[4] = {-1.f, 1.f, 1.f, -1.f};
    const float dys[4] = {-1.f, -1.f, 1.f, 1.f};

    float c1x[4], c1y[4], c2x[4], c2y[4];
#pragma unroll
    for (int k = 0; k < 4; ++k) {
      const float ax = dxs[k] * pw * 0.5f, ay = dys[k] * ph * 0.5f;
      c1x[k] = px + ax * c1a - ay * s1;
      c1y[k] = py + ax * s1 + ay * c1a;
      const float bx = dxs[k] * qw * 0.5f, by = dys[k] * qh * 0.5f;
      c2x[k] = qx + bx * c2a - by * s2;
      c2y[k] = qy + bx * s2 + by * c2a;
    }

    float X[24], Y[24];
    bool  Mk[24];

    // 16 edge-edge intersection candidates (fixed slots 0..15)
#pragma unroll
    for (int e1 = 0; e1 < 4; ++e1) {
      const float p1x = c1x[e1], p1y = c1y[e1];
      const float rx = c1x[(e1 + 1) & 3] - p1x;
      const float ry = c1y[(e1 + 1) & 3] - p1y;
#pragma unroll
      for (int e2 = 0; e2 < 4; ++e2) {
        const float q1x = c2x[e2], q1y = c2y[e2];
        const float sx = c2x[(e2 + 1) & 3] - q1x;
        const float sy = c2y[(e2 + 1) & 3] - q1y;
        const float den = rx * sy - ry * sx;
        const bool dok = fabsf(den) > DEN_EPS;
        const float dsafe = dok ? den : 1.0f;
        const float qpx = q1x - p1x, qpy = q1y - p1y;
        const float t = (qpx * sy - qpy * sx) / dsafe;
        const float u = (qpx * ry - qpy * rx) / dsafe;
        const bool ok = dok && (t >= 0.f) && (t <= 1.f) && (u >= 0.f) && (u <= 1.f);
        const float ts = ok ? t : 0.0f;
        const int slot = e1 * 4 + e2;
        X[slot] = p1x + ts * rx;
        Y[slot] = p1y + ts * ry;
        Mk[slot] = ok;
      }
    }

    // pred corners inside target (slots 16..19)
#pragma unroll
    for (int k = 0; k < 4; ++k) {
      const float dx = c1x[k] - qx, dy = c1y[k] - qy;
      const float lx =  dx * c2a + dy * s2;
      const float ly = -dx * s2  + dy * c2a;
      X[16 + k] = c1x[k]; Y[16 + k] = c1y[k];
      Mk[16 + k] = (fabsf(lx) <= qw * 0.5f + IN_TOL) &&
                   (fabsf(ly) <= qh * 0.5f + IN_TOL);
    }
    // target corners inside pred (slots 20..23)
#pragma unroll
    for (int k = 0; k < 4; ++k) {
      const float dx = c2x[k] - px, dy = c2y[k] - py;
      const float lx =  dx * c1a + dy * s1;
      const float ly = -dx * s1  + dy * c1a;
      X[20 + k] = c2x[k]; Y[20 + k] = c2y[k];
      Mk[20 + k] = (fabsf(lx) <= pw * 0.5f + IN_TOL) &&
                   (fabsf(ly) <= ph * 0.5f + IN_TOL);
    }

    int num = 0;
    float sx_ = 0.f, sy_ = 0.f;
#pragma unroll
    for (int j = 0; j < 24; ++j) {
      num += Mk[j] ? 1 : 0;
      sx_ += Mk[j] ? X[j] : 0.f;
      sy_ += Mk[j] ? Y[j] : 0.f;
    }
    const float numf = (float)(num > 1 ? num : 1);
    const float cx = sx_ / numf, cy = sy_ / numf;

    // Pseudo-angle: strictly monotone in atan2(dy,dx) over (-pi, pi], range
    // [-2, 2]. Identical sort order => identical polygon (up to a cyclic
    // rotation, which leaves the shoelace area unchanged). ~8 ops vs ~25+
    // for atan2f. Invalid slots get +1e8 so they sort to the tail.
    float ang[24];
#pragma unroll
    for (int j = 0; j < 24; ++j) {
      const float dx = X[j] - cx, dy = Y[j] - cy;
      const float ad = fabsf(dx) + fabsf(dy);
      const float r = dx / ad;                       // NaN if ad==0 (guarded)
      const float pa_ = (dy >= 0.f) ? (1.f - r) : (r - 1.f);
      ang[j] = Mk[j] ? ((ad > 0.f) ? pa_ : 0.f) : 1.0e8f;
    }

    // Batcher odd-even mergesort (<=191 CEs vs 276 for transposition sort);
    // constexpr indices keep everything in registers.
#pragma unroll
    for (int c = 0; c < SORT_NET.cnt; ++c) {
      const int lo = SORT_NET.lo[c];
      const int hi = SORT_NET.hi[c];
      const bool sw = ang[lo] > ang[hi];
      const float ta = ang[lo], tx = X[lo], ty = Y[lo];
      ang[lo] = sw ? ang[hi] : ta;
      X[lo]   = sw ? X[hi]   : tx;
      Y[lo]   = sw ? Y[hi]   : ty;
      ang[hi] = sw ? ta : ang[hi];
      X[hi]   = sw ? tx : X[hi];
      Y[hi]   = sw ? ty : Y[hi];
    }

    // invalid (angle=1e8) sorted to tail; replace with first sorted point
    const float x0 = X[0], y0 = Y[0];
    float Xs[24], Ys[24];
#pragma unroll
    for (int j = 0; j < 24; ++j) {
      const bool v = j < num;
      Xs[j] = v ? X[j] : x0;
      Ys[j] = v ? Y[j] : y0;
    }
    float ssum = 0.f;
#pragma unroll
    for (int j = 0; j < 24; ++j) {
      const int jn = (j + 1) % 24;  // compile-time after unroll
      ssum += Xs[j] * Ys[jn] - Ys[j] * Xs[jn];
    }
    float area_i = (num >= 3) ? 0.5f * fabsf(ssum) : 0.0f;

    const float a1 = pw * ph, a2 = qw * qh;
    const float iou = area_i / fmaxf(a1 + a2 - area_i, DEN_EPS);
    loss = -logf(fmaxf(iou, LOSS_EPS));
  }

  // wave32 tree reduction, then cross-wave via LDS (deterministic order)
#pragma unroll
  for (int off = 16; off > 0; off >>= 1)
    loss += __shfl_down(loss, off, 32);

  __shared__ float wsum[8];
  if ((threadIdx.x & 31) == 0) wsum[threadIdx.x >> 5] = loss;
  __syncthreads();
  if (threadIdx.x == 0) {
    float s = 0.f;
#pragma unroll
    for (int k = 0; k < 8; ++k) s += wsum[k];
    partial[blockIdx.x] = s;
  }
}

// ---------------------------------------------------------------------------
// Kernel 2: single-wave final reduction using V_WMMA_F32_16X16X4_F32 as a
// 64-way sum engine: D = A(16x4 chunk of partials) * B(ones) + C(accumulator).
// EXEC stays all-ones: 32 threads, uniform trip count, unconditional loads
// (clamped index + masked multiply for the tail).
// ---------------------------------------------------------------------------
__global__ __launch_bounds__(32) void final_reduce_wmma(
    const float* __restrict__ part, int nb, float inv_n, float* __restrict__ out) {
  const int lane = threadIdx.x;          // 0..31
  const int m = lane & 15;               // A-matrix row
  const int k0 = (lane >> 4) * 2;        // A-matrix K base (VGPR0/VGPR1 = k0, k0+1)

  v8f acc = {0.f, 0.f, 0.f, 0.f, 0.f, 0.f, 0.f, 0.f};
  v2f bones; bones[0] = 1.0f; bones[1] = 1.0f;  // B = 4x16 all-ones

  for (int base = 0; base < nb; base += 64) {
    const int i0 = base + m * 4 + k0;
    const int i1 = i0 + 1;
    const float a0 = part[i0 < nb ? i0 : nb - 1] * (i0 < nb ? 1.0f : 0.0f);
    const float a1 = part[i1 < nb ? i1 : nb - 1] * (i1 < nb ? 1.0f : 0.0f);
    v2f a; a[0] = a0; a[1] = a1;
    // D[m][n] = sum_k A[m][k]*1 + C[m][n]  -> 16 persistent row accumulators
    acc = __builtin_amdgcn_wmma_f32_16x16x4_f32(
        false, a, false, bones, (short)0, acc, false, false);
  }

  // D layout: VGPR r, lanes 0-15 hold M=r; lanes 16-31 hold M=8+r (row sums
  // replicated across N). Fold 8 VGPRs, then combine lane 0 (rows 0-7) with
  // lane 16 (rows 8-15).
  float s = acc[0] + acc[1] + acc[2] + acc[3] + acc[4] + acc[5] + acc[6] + acc[7];
  const float total = __shfl(s, 0, 32) + __shfl(s, 16, 32);
  if (lane == 0) out[0] = total * inv_n;
}

extern "C" void kernel_launch(void* const* d_in, const int* in_sizes, int n_in,
                              void* d_out, int out_size, void* d_ws, size_t ws_size,
                              hipStream_t stream) {
  const float* pred   = (const float*)d_in[0];
  const float* target = (const float*)d_in[1];
  const int n  = in_sizes[0] / 5;
  const int nb = (n + 255) / 256;

  float* partial = (float*)d_ws;  // nb floats of scratch (fully written below)

  rot_iou_loss_kernel<<<nb, 256, 0, stream>>>(pred, target, partial, n);
  final_reduce_wmma<<<1, 32, 0, stream>>>(partial, nb, 1.0f / (float)n,
                                          (float*)d_out);
}